// PPGEncoder_57904749084988
// MI455X (gfx1250) — compile-verified
//
#include <hip/hip_runtime.h>
#include <hip/hip_bf16.h>
#include <math.h>

// CDNA5 / gfx1250 implementation of PPGEncoder.
// Roofline: output 64x512x4096 f32 = 512MB -> ~22us at 23.3 TB/s; total GEMM
// FLOPs ~3G -> compute-free. All three MLP GEMMs use V_WMMA_F32_16X16X4_F32
// (f32 precision preserved; the job is store-bound so lower-precision WMMA
// buys nothing). GEMM3's 512MB result uses nontemporal stores so the stream
// doesn't thrash the 192MB L2 that holds the (re-used) weights/activations.
// Workspace requirement: ~2.84 MB.

typedef __attribute__((ext_vector_type(2))) float v2f;
typedef __attribute__((ext_vector_type(8))) float v8f;

#define B_DEF 64
#define C_IN 4
#define L_SEQ 256
#define H_DIM 4
#define N_STATE 4

// ===================================================================
// S4 front-end (tiny, VALU kernels)
// ===================================================================

// u[b,l,h] = sum_c x[b,c,l]*enc_W[c,h] + enc_b[h]
__global__ void k_encoder(const float* __restrict__ x, const float* __restrict__ W,
                          const float* __restrict__ bias, float* __restrict__ u) {
    int t = blockIdx.x * blockDim.x + threadIdx.x;   // t = b*256 + l
    int b = t >> 8, l = t & 255;
    float xs[C_IN];
#pragma unroll
    for (int c = 0; c < C_IN; ++c) xs[c] = x[(b * C_IN + c) * L_SEQ + l];
#pragma unroll
    for (int h = 0; h < H_DIM; ++h) {
        float acc = bias[h];
#pragma unroll
        for (int c = 0; c < C_IN; ++c) acc += xs[c] * W[c * H_DIM + h];
        u[t * H_DIM + h] = acc;
    }
}

// SSM kernel K[h,l] = 2*Re( sum_n C[h,n]*Bd[h,n]*exp(dtA[h,n]*l) )
__global__ void k_ssm_kernel(const float* __restrict__ log_dt, const float* __restrict__ A_re,
                             const float* __restrict__ A_im, const float* __restrict__ C_re,
                             const float* __restrict__ C_im, float* __restrict__ Kout) {
    int idx = blockIdx.x * blockDim.x + threadIdx.x;  // h*256 + l
    int h = idx >> 8;
    float l = (float)(idx & 255);
    float dt = __expf(log_dt[h]);
    float acc = 0.0f;
#pragma unroll
    for (int n = 0; n < N_STATE; ++n) {
        float Are = -__expf(A_re[h * N_STATE + n]);
        float Aim = A_im[h * N_STATE + n];
        float dr = Are * dt, di = Aim * dt;
        float e = __expf(dr);
        float er = e * __cosf(di) - 1.0f;
        float ei = e * __sinf(di);
        float den = Are * Are + Aim * Aim;
        float Br = (er * Are + ei * Aim) / den;
        float Bi = (ei * Are - er * Aim) / den;
        float Cr = C_re[h * N_STATE + n], Ci = C_im[h * N_STATE + n];
        float CBr = Cr * Br - Ci * Bi;
        float CBi = Cr * Bi + Ci * Br;
        float vr = __expf(dr * l);
        float vre = vr * __cosf(di * l), vim = vr * __sinf(di * l);
        acc += CBr * vre - CBi * vim;
    }
    Kout[idx] = 2.0f * acc;
}

// prenorm over H=4, write transposed z[b,h,l]
__global__ void k_prenorm(const float* __restrict__ u, const float* __restrict__ g,
                          const float* __restrict__ be, float* __restrict__ z) {
    int t = blockIdx.x * blockDim.x + threadIdx.x;   // b*256 + l
    int b = t >> 8, l = t & 255;
    float v[H_DIM];
    float mu = 0.0f;
#pragma unroll
    for (int h = 0; h < H_DIM; ++h) { v[h] = u[t * H_DIM + h]; mu += v[h]; }
    mu *= 0.25f;
    float var = 0.0f;
#pragma unroll
    for (int h = 0; h < H_DIM; ++h) { float d = v[h] - mu; var += d * d; }
    var *= 0.25f;
    float rinv = rsqrtf(var + 1e-5f);
#pragma unroll
    for (int h = 0; h < H_DIM; ++h)
        z[(b * H_DIM + h) * L_SEQ + l] = (v[h] - mu) * rinv * g[h] + be[h];
}

// causal conv + D skip + GELU(tanh approx).  One block per (b,h).
__global__ void k_conv_gelu(const float* __restrict__ z, const float* __restrict__ Kk,
                            const float* __restrict__ D, float* __restrict__ gy) {
    __shared__ float sK[L_SEQ];
    __shared__ float sz[L_SEQ];
    int bh = blockIdx.x;                 // b*H + h
    int h = bh & (H_DIM - 1);
    int l = threadIdx.x;
    sK[l] = Kk[h * L_SEQ + l];
    sz[l] = z[bh * L_SEQ + l];
    __syncthreads();
    float acc = 0.0f;
    for (int j = 0; j <= l; ++j) acc += sK[j] * sz[l - j];
    float y = acc + D[h] * sz[l];
    float y3 = y * y * y;
    float t = tanhf(0.7978845608028654f * (y + 0.044715f * y3));
    gy[bh * L_SEQ + l] = 0.5f * y * (1.0f + t);
}

// u[b,l,k] += sum_h gy[b,h,l]*out_W[h,k] + out_b[k]
__global__ void k_block_out(const float* __restrict__ gy, const float* __restrict__ W,
                            const float* __restrict__ bias, float* __restrict__ u) {
    int t = blockIdx.x * blockDim.x + threadIdx.x;   // b*256 + l
    int b = t >> 8, l = t & 255;
    float g[H_DIM];
#pragma unroll
    for (int h = 0; h < H_DIM; ++h) g[h] = gy[(b * H_DIM + h) * L_SEQ + l];
#pragma unroll
    for (int k = 0; k < H_DIM; ++k) {
        float acc = bias[k];
#pragma unroll
        for (int h = 0; h < H_DIM; ++h) acc += g[h] * W[h * H_DIM + k];
        u[t * H_DIM + k] += acc;
    }
}

// s4out[b,l,h] = sum_j u[b,l,j]*dec_W[j,h] + dec_b[h]
__global__ void k_decoder(const float* __restrict__ u, const float* __restrict__ W,
                          const float* __restrict__ bias, float* __restrict__ s4out) {
    int t = blockIdx.x * blockDim.x + threadIdx.x;
    float v[H_DIM];
#pragma unroll
    for (int j = 0; j < H_DIM; ++j) v[j] = u[t * H_DIM + j];
#pragma unroll
    for (int h = 0; h < H_DIM; ++h) {
        float acc = bias[h];
#pragma unroll
        for (int j = 0; j < H_DIM; ++j) acc += v[j] * W[j * H_DIM + h];
        s4out[t * H_DIM + h] = acc;
    }
}

// combined[b,1856] = [z16 (64) | per-c: pool64(64)+pool128(128)+pool256(256)]
__global__ void k_combined(const float* __restrict__ s4out, const float* __restrict__ x,
                           float* __restrict__ comb) {
    __shared__ float sx[L_SEQ];
    int b = blockIdx.x;
    int t = threadIdx.x;
    float* cb = comb + b * 1856;
    if (t < 64) {
        int h = t >> 4, o = t & 15;
        float acc = 0.0f;
#pragma unroll
        for (int j = 0; j < 16; ++j)
            acc += s4out[((b * L_SEQ) + (o * 16 + j)) * H_DIM + h];
        cb[t] = acc * (1.0f / 16.0f);
    }
    for (int c = 0; c < C_IN; ++c) {
        sx[t] = x[(b * C_IN + c) * L_SEQ + t];
        __syncthreads();
        float* base = cb + 64 + c * 448;
        base[192 + t] = sx[t];
        if (t < 128) base[64 + t] = 0.5f * (sx[2 * t] + sx[2 * t + 1]);
        if (t < 64)
            base[t] = 0.25f * (sx[4 * t] + sx[4 * t + 1] + sx[4 * t + 2] + sx[4 * t + 3]);
        __syncthreads();
    }
}

// ===================================================================
// f32 WMMA GEMM (compile-time shapes): C[M,N] = A[M,K] @ B[K,N] + bias[N]
// One wave per 16x16 tile; K stepped by 4; unroll-by-4 -> 4 WMMA per body
// with strength-reduced pointer bumps and one B-prefetch per 16 K.
// ===================================================================
template <int M, int N, int K>
__global__ void k_gemm_wmma_f32(const float* __restrict__ A, const float* __restrict__ B,
                                const float* __restrict__ bias, float* __restrict__ C) {
    constexpr int NT = N / 16;
    int wave = (blockIdx.x * blockDim.x + threadIdx.x) >> 5;
    int lane = threadIdx.x & 31;
    int m0 = (wave / NT) << 4;
    int n0 = (wave % NT) << 4;
    int lm = lane & 15;                 // M (A) / N (B,C) within tile
    int koff = (lane >> 4) << 1;        // lanes 16-31 hold K+2,K+3

    float bval = bias[n0 + lm];
    v8f c;
#pragma unroll
    for (int r = 0; r < 8; ++r) c[r] = bval;

    const float* Ap = A + (m0 + lm) * K + koff;
    const float* Bp = B + koff * N + n0 + lm;
#pragma unroll 4
    for (int k0 = 0; k0 < K; k0 += 4) {
        if ((k0 & 12) == 0 && k0 + 16 < K)          // 1 prefetch per 16 K
            __builtin_prefetch(Bp + 16 * N, 0, 1);  // global_prefetch_b8
        v2f a, bb;
        a.x = Ap[0];
        a.y = Ap[1];
        bb.x = Bp[0];
        bb.y = Bp[N];
        c = __builtin_amdgcn_wmma_f32_16x16x4_f32(false, a, false, bb,
                                                  (short)0, c, false, false);
        Ap += 4;
        Bp += 4 * N;
    }
    int row = m0 + ((lane >> 4) << 3);  // lanes 16-31 write rows +8..+15
#pragma unroll
    for (int r = 0; r < 8; ++r)
        C[(row + r) * N + n0 + lm] = c[r];
}

// ===================================================================
// GEMM3 (store-bound): A=[32768,8], B=W3[8,4096], out 512MB f32.
// One wave per 16x64 strip: A fragment loaded once, 4 n-tiles, 2 WMMA each,
// nontemporal stores (output never re-read; keep L2 for resident weights).
// ===================================================================
__global__ void k_gemm3_wmma(const float* __restrict__ A, const float* __restrict__ B,
                             const float* __restrict__ bias, float* __restrict__ C) {
    constexpr int N = 4096, K = 8;
    int wave = (blockIdx.x * blockDim.x + threadIdx.x) >> 5;
    int lane = threadIdx.x & 31;
    int m0 = (wave >> 6) << 4;          // 2048 m-tiles
    int ns = (wave & 63) << 6;          // 64 strips of 64 columns
    int lm = lane & 15;
    int koff = (lane >> 4) << 1;

    const float* Ap = A + (m0 + lm) * K + koff;
    float a0 = Ap[0], a1 = Ap[1], a2 = Ap[4], a3 = Ap[5];
    int row = m0 + ((lane >> 4) << 3);

#pragma unroll
    for (int t = 0; t < 4; ++t) {
        int n0 = ns + t * 16;
        float bval = bias[n0 + lm];
        v8f c;
#pragma unroll
        for (int r = 0; r < 8; ++r) c[r] = bval;
        const float* Bp = B + koff * N + n0 + lm;
        v2f a, bb;
        a.x = a0; a.y = a1;
        bb.x = Bp[0]; bb.y = Bp[N];
        c = __builtin_amdgcn_wmma_f32_16x16x4_f32(false, a, false, bb,
                                                  (short)0, c, false, false);
        a.x = a2; a.y = a3;
        Bp += 4 * N;
        bb.x = Bp[0]; bb.y = Bp[N];
        c = __builtin_amdgcn_wmma_f32_16x16x4_f32(false, a, false, bb,
                                                  (short)0, c, false, false);
#pragma unroll
        for (int r = 0; r < 8; ++r)
            __builtin_nontemporal_store(c[r], C + (size_t)(row + r) * N + n0 + lm);
    }
}

// row-wise LayerNorm + ReLU, in place.  One block per row.
__global__ void k_ln_relu(float* __restrict__ h, const float* __restrict__ g,
                          const float* __restrict__ be, int N) {
    __shared__ float ssum[256];
    __shared__ float ssq[256];
    int tid = threadIdx.x;
    float* row = h + (size_t)blockIdx.x * N;
    float s = 0.0f, q = 0.0f;
    for (int i = tid; i < N; i += 256) { float v = row[i]; s += v; q += v * v; }
    ssum[tid] = s; ssq[tid] = q;
    __syncthreads();
    for (int o = 128; o > 0; o >>= 1) {
        if (tid < o) { ssum[tid] += ssum[tid + o]; ssq[tid] += ssq[tid + o]; }
        __syncthreads();
    }
    float mu = ssum[0] / (float)N;
    float var = ssq[0] / (float)N - mu * mu;
    float rinv = rsqrtf(var + 1e-5f);
    for (int i = tid; i < N; i += 256) {
        float v = (row[i] - mu) * rinv * g[i] + be[i];
        row[i] = fmaxf(v, 0.0f);
    }
}

// ===================================================================
extern "C" void kernel_launch(void* const* d_in, const int* in_sizes, int n_in,
                              void* d_out, int out_size, void* d_ws, size_t ws_size,
                              hipStream_t stream) {
    const float* x     = (const float*)d_in[0];
    const float* enc_W = (const float*)d_in[1];
    const float* enc_b = (const float*)d_in[2];
    const float* ln_s  = (const float*)d_in[3];
    const float* ln_b  = (const float*)d_in[4];
    const float* log_dt= (const float*)d_in[5];
    const float* A_re  = (const float*)d_in[6];
    const float* A_im  = (const float*)d_in[7];
    const float* C_re  = (const float*)d_in[8];
    const float* C_im  = (const float*)d_in[9];
    const float* Dp    = (const float*)d_in[10];
    const float* out_W = (const float*)d_in[11];
    const float* out_b = (const float*)d_in[12];
    const float* dec_W = (const float*)d_in[13];
    const float* dec_b = (const float*)d_in[14];
    const float* W1    = (const float*)d_in[15];
    const float* b1    = (const float*)d_in[16];
    const float* g1    = (const float*)d_in[17];
    const float* be1   = (const float*)d_in[18];
    const float* W2    = (const float*)d_in[19];
    const float* b2    = (const float*)d_in[20];
    const float* g2    = (const float*)d_in[21];
    const float* be2   = (const float*)d_in[22];
    const float* W3    = (const float*)d_in[23];
    const float* b3    = (const float*)d_in[24];
    float* out = (float*)d_out;

    // workspace layout (floats), total 709632 f = 2.84MB
    float* ws = (float*)d_ws;
    float* u        = ws;                 // 65536
    float* z        = ws + 65536;         // 65536
    float* gy       = ws + 131072;        // 65536
    float* Kbuf     = ws + 196608;        // 1024
    float* s4out    = ws + 197632;        // 65536
    float* combined = ws + 263168;        // 118784
    float* h1       = ws + 381952;        // 65536
    float* h2       = ws + 447488;        // 262144

    k_encoder<<<64, 256, 0, stream>>>(x, enc_W, enc_b, u);

    for (int blk = 0; blk < 2; ++blk) {
        k_ssm_kernel<<<4, 256, 0, stream>>>(log_dt + blk * H_DIM,
                                            A_re + blk * H_DIM * N_STATE,
                                            A_im + blk * H_DIM * N_STATE,
                                            C_re + blk * H_DIM * N_STATE,
                                            C_im + blk * H_DIM * N_STATE, Kbuf);
        k_prenorm<<<64, 256, 0, stream>>>(u, ln_s + blk * H_DIM, ln_b + blk * H_DIM, z);
        k_conv_gelu<<<B_DEF * H_DIM, L_SEQ, 0, stream>>>(z, Kbuf, Dp + blk * H_DIM, gy);
        k_block_out<<<64, 256, 0, stream>>>(gy, out_W + blk * H_DIM * H_DIM,
                                            out_b + blk * H_DIM, u);
    }

    k_decoder<<<64, 256, 0, stream>>>(u, dec_W, dec_b, s4out);
    k_combined<<<B_DEF, 256, 0, stream>>>(s4out, x, combined);

    // GEMM1: [64,1856]x[1856,1024] -> 256 tiles -> 32 blocks of 8 waves
    k_gemm_wmma_f32<64, 1024, 1856><<<32, 256, 0, stream>>>(combined, W1, b1, h1);
    k_ln_relu<<<64, 256, 0, stream>>>(h1, g1, be1, 1024);

    // GEMM2: [64,1024]x[1024,4096] -> 1024 tiles -> 128 blocks
    k_gemm_wmma_f32<64, 4096, 1024><<<128, 256, 0, stream>>>(h1, W2, b2, h2);
    k_ln_relu<<<64, 256, 0, stream>>>(h2, g2, be2, 4096);

    // GEMM3: 2048 m-tiles x 64 strips = 131072 waves -> 16384 blocks
    k_gemm3_wmma<<<16384, 256, 0, stream>>>(h2, W3, b3, out);
}